// Decoder_16758962389695
// MI455X (gfx1250) — compile-verified
//
#include <hip/hip_runtime.h>
#include <hip/hip_bf16.h>

#define NS 16384   // 128*128 spatial
#define C1 256
#define C2 512

typedef float v2f __attribute__((ext_vector_type(2)));
typedef float v8f __attribute__((ext_vector_type(8)));

__device__ __forceinline__ v8f wmma_f32(v2f a, v2f b, v8f c) {
  // V_WMMA_F32_16X16X4_F32 : D = A(16x4) * B(4x16) + C(16x16), fp32 exact
  return __builtin_amdgcn_wmma_f32_16x16x4_f32(false, a, false, b, (short)0, c,
                                               false, false);
}

// ---------------------------------------------------------------- upsample
// align_corners=True bilinear 32x32 -> 128x128, writes channels [0,256) of cc
__global__ void k_upsample(const float* __restrict__ x, float* __restrict__ cc) {
  int i = blockIdx.x * 256 + threadIdx.x;            // 256*16384
  if (i >= C1 * NS) return;
  int c = i / NS, n = i % NS, oy = n >> 7, ox = n & 127;
  const float s = 31.0f / 127.0f;
  float sy = oy * s, sx = ox * s;
  int y0 = (int)floorf(sy); int y1 = (y0 + 1 > 31) ? 31 : y0 + 1; float wy = sy - (float)y0;
  int x0 = (int)floorf(sx); int x1 = (x0 + 1 > 31) ? 31 : x0 + 1; float wx = sx - (float)x0;
  const float* xc = x + c * 1024;
  float a00 = xc[y0 * 32 + x0], a01 = xc[y0 * 32 + x1];
  float a10 = xc[y1 * 32 + x0], a11 = xc[y1 * 32 + x1];
  float r0 = a00 * (1.f - wy) + a10 * wy;
  float r1 = a01 * (1.f - wy) + a11 * wy;
  cc[i] = r0 * (1.f - wx) + r1 * wx;
}

// ------------------------------------------------------------- 1x1 conv GEMM
// D[m][n] = sum_k A[m][k]*B[k][n], M=K=256, N=16384; 2x2 wave tiles (32x32)
__global__ void k_gemm1x1(const float* __restrict__ A, const float* __restrict__ B,
                          float* __restrict__ D, int applySigmoid) {
  int lane = threadIdx.x & 31, wv = threadIdx.x >> 5;
  int t = blockIdx.x * 8 + wv;                       // 8 mt * 512 nt
  int mt = t >> 9, nt = t & 511;
  int m0 = mt * 32, n0 = nt * 32;
  int ln = lane & 15, lh = lane >> 4;
  const float* Ar0 = A + (size_t)(m0 + ln) * 256;
  const float* Ar1 = A + (size_t)(m0 + 16 + ln) * 256;
  const float* Bc0 = B + n0 + ln;
  const float* Bc1 = B + n0 + 16 + ln;
  v8f acc00 = {}, acc01 = {}, acc10 = {}, acc11 = {};
  for (int k = 0; k < 256; k += 4) {
    int ka = k + 2 * lh;
    v2f a0, a1, b0, b1;
    a0.x = Ar0[ka]; a0.y = Ar0[ka + 1];
    a1.x = Ar1[ka]; a1.y = Ar1[ka + 1];
    b0.x = Bc0[(size_t)ka * NS]; b0.y = Bc0[(size_t)(ka + 1) * NS];
    b1.x = Bc1[(size_t)ka * NS]; b1.y = Bc1[(size_t)(ka + 1) * NS];
    acc00 = wmma_f32(a0, b0, acc00);
    acc01 = wmma_f32(a0, b1, acc01);
    acc10 = wmma_f32(a1, b0, acc10);
    acc11 = wmma_f32(a1, b1, acc11);
  }
#pragma unroll
  for (int r = 0; r < 8; ++r) {
    int mA = m0 + r + 8 * lh, mB = mA + 16;
    float v00 = acc00[r], v01 = acc01[r], v10 = acc10[r], v11 = acc11[r];
    if (applySigmoid) {
      v00 = 1.0f / (1.0f + expf(-v00));
      v01 = 1.0f / (1.0f + expf(-v01));
      v10 = 1.0f / (1.0f + expf(-v10));
      v11 = 1.0f / (1.0f + expf(-v11));
    }
    D[(size_t)mA * NS + n0 + ln]      = v00;
    D[(size_t)mA * NS + n0 + 16 + ln] = v01;
    D[(size_t)mB * NS + n0 + ln]      = v10;
    D[(size_t)mB * NS + n0 + 16 + ln] = v11;
  }
}

// ---------------------------------------------- per-channel 128x128 matmuls
// u[c] = L[c] @ v[c] + L[c] ; 2x2 wave tiles, residual seeded into acc
__global__ void k_chanmm(const float* __restrict__ L, const float* __restrict__ V,
                         float* __restrict__ U) {
  int lane = threadIdx.x & 31, wv = threadIdx.x >> 5;
  int t = blockIdx.x * 8 + wv;                       // 256 ch * (4 mt * 4 nt)
  int c = t >> 4, ti = t & 15, mt = ti >> 2, nt = ti & 3;
  int m0 = mt * 32, n0 = nt * 32;
  int ln = lane & 15, lh = lane >> 4;
  const float* Lc = L + (size_t)c * NS;
  const float* Vc = V + (size_t)c * NS;
  v8f acc00, acc01, acc10, acc11;
#pragma unroll
  for (int r = 0; r < 8; ++r) {
    int mA = m0 + r + 8 * lh, mB = mA + 16;
    acc00[r] = Lc[mA * 128 + n0 + ln];
    acc01[r] = Lc[mA * 128 + n0 + 16 + ln];
    acc10[r] = Lc[mB * 128 + n0 + ln];
    acc11[r] = Lc[mB * 128 + n0 + 16 + ln];
  }
  for (int k = 0; k < 128; k += 4) {
    int ka = k + 2 * lh;
    v2f a0, a1, b0, b1;
    a0.x = Lc[(m0 + ln) * 128 + ka];      a0.y = Lc[(m0 + ln) * 128 + ka + 1];
    a1.x = Lc[(m0 + 16 + ln) * 128 + ka]; a1.y = Lc[(m0 + 16 + ln) * 128 + ka + 1];
    b0.x = Vc[ka * 128 + n0 + ln];        b0.y = Vc[(ka + 1) * 128 + n0 + ln];
    b1.x = Vc[ka * 128 + n0 + 16 + ln];   b1.y = Vc[(ka + 1) * 128 + n0 + 16 + ln];
    acc00 = wmma_f32(a0, b0, acc00);
    acc01 = wmma_f32(a0, b1, acc01);
    acc10 = wmma_f32(a1, b0, acc10);
    acc11 = wmma_f32(a1, b1, acc11);
  }
  float* Uc = U + (size_t)c * NS;
#pragma unroll
  for (int r = 0; r < 8; ++r) {
    int mA = m0 + r + 8 * lh, mB = mA + 16;
    Uc[mA * 128 + n0 + ln]      = acc00[r];
    Uc[mA * 128 + n0 + 16 + ln] = acc01[r];
    Uc[mB * 128 + n0 + ln]      = acc10[r];
    Uc[mB * 128 + n0 + 16 + ln] = acc11[r];
  }
}

// ------------------------------------------- offset conv (3x3, 512ch -> 18)
// implicit im2col GEMM, M padded 18->32 with guards; K = 512*9, k = c*9 + r
__global__ void k_offconv(const float* __restrict__ cc, const float* __restrict__ pw,
                          const float* __restrict__ pb, float* __restrict__ off) {
  int lane = threadIdx.x & 31, wv = threadIdx.x >> 5;
  int t = blockIdx.x * 8 + wv;                       // 2 mt * 1024 nt
  int mt = t >> 10, nt = t & 1023;
  int m0 = mt * 16, n0 = nt * 16;
  int ln = lane & 15, lh = lane >> 4;
  int n = n0 + ln, h = n >> 7, w = n & 127;
  bool mvalid = (m0 + ln) < 18;
  const float* Ar = pw + (size_t)(m0 + ln) * 4608;
  v8f acc = {};
  for (int k = 0; k < 4608; k += 4) {
    int ka = k + 2 * lh;
    v2f a, b;
    a.x = mvalid ? Ar[ka] : 0.f;
    a.y = mvalid ? Ar[ka + 1] : 0.f;
#pragma unroll
    for (int vv = 0; vv < 2; ++vv) {
      int kk = ka + vv, c = kk / 9, r = kk - c * 9;
      int hh = h + r / 3 - 1, ww = w + (r % 3) - 1;
      float bv = 0.f;
      if (hh >= 0 && hh < 128 && ww >= 0 && ww < 128)
        bv = cc[(size_t)c * NS + hh * 128 + ww];
      if (vv == 0) b.x = bv; else b.y = bv;
    }
    acc = wmma_f32(a, b, acc);
  }
#pragma unroll
  for (int r = 0; r < 8; ++r) {
    int m = m0 + r + 8 * lh;
    if (m < 18) off[(size_t)m * NS + n] = acc[r] + pb[m];
  }
}

// ------------------------------------------ bilinear gather tables for DCN
__global__ void k_dprep(const float* __restrict__ off, int* __restrict__ idx4,
                        float* __restrict__ w4) {
  int i = blockIdx.x * 256 + threadIdx.x;            // NS * 9
  if (i >= NS * 9) return;
  int n = i / 9, kp = i - n * 9;
  int h = n >> 7, w = n & 127;
  float px = off[(size_t)kp * NS + n] + (float)(kp / 3 - 1) + (float)(h + 1);
  float py = off[(size_t)(9 + kp) * NS + n] + (float)(kp % 3 - 1) + (float)(w + 1);
  float fx = floorf(px), fy = floorf(py);
  float x0 = fminf(fmaxf(fx,        0.f), 129.f);
  float x1 = fminf(fmaxf(fx + 1.f,  0.f), 129.f);
  float y0 = fminf(fmaxf(fy,        0.f), 129.f);
  float y1 = fminf(fmaxf(fy + 1.f,  0.f), 129.f);
  float pxc = fminf(fmaxf(px, 0.f), 129.f);
  float pyc = fminf(fmaxf(py, 0.f), 129.f);
  float glt = (1.f + (x0 - pxc)) * (1.f + (y0 - pyc));
  float grb = (1.f - (x1 - pxc)) * (1.f - (y1 - pyc));
  float glb = (1.f + (x0 - pxc)) * (1.f - (y1 - pyc));
  float grt = (1.f - (x1 - pxc)) * (1.f + (y0 - pyc));
  int   xi[4] = {(int)x0, (int)x1, (int)x0, (int)x1};
  int   yi[4] = {(int)y0, (int)y1, (int)y1, (int)y0};
  float gw[4] = {glt, grb, glb, grt};
  int base = i * 4;
#pragma unroll
  for (int j = 0; j < 4; ++j) {
    int X = xi[j], Y = yi[j];
    bool valid = (X >= 1 && X <= 128 && Y >= 1 && Y <= 128);
    idx4[base + j] = valid ? ((X - 1) * 128 + (Y - 1)) : 0;
    w4[base + j]   = valid ? gw[j] : 0.f;
  }
}

// ---------------------------------------------------- deform-conv gather-GEMM
// fam[o][n] = Cf[o][n] + sum_{r,c} dc_w[o][c][r] * bilinear(cc[c], n, r)
// K ordered k = r*512 + c so (idx,w) stay in registers across 512-ch inner loop;
// 2x2 wave tiles so each gathered B fragment feeds two M-tiles.
__global__ void k_deform(const float* __restrict__ cc, const float* __restrict__ dcw,
                         const int* __restrict__ idx4, const float* __restrict__ w4,
                         float* __restrict__ fam) {
  int lane = threadIdx.x & 31, wv = threadIdx.x >> 5;
  int t = blockIdx.x * 8 + wv;                       // 8 mt * 512 nt
  int mt = t >> 9, nt = t & 511;
  int m0 = mt * 32, n0 = nt * 32;
  int ln = lane & 15, lh = lane >> 4;
  int nA = n0 + ln, nB = n0 + 16 + ln;
  v8f acc00, acc01, acc10, acc11;
#pragma unroll
  for (int r = 0; r < 8; ++r) {      // residual: Cf lives in cc channels [256,512)
    int mA = m0 + r + 8 * lh, mB = mA + 16;
    acc00[r] = cc[(size_t)(C1 + mA) * NS + nA];
    acc01[r] = cc[(size_t)(C1 + mA) * NS + nB];
    acc10[r] = cc[(size_t)(C1 + mB) * NS + nA];
    acc11[r] = cc[(size_t)(C1 + mB) * NS + nB];
  }
  const float* Am0 = dcw + (size_t)(m0 + ln) * 4608;       // [m][c][rp] -> Am[c*9+rp]
  const float* Am1 = dcw + (size_t)(m0 + 16 + ln) * 4608;
  for (int rp = 0; rp < 9; ++rp) {
    int wa = (nA * 9 + rp) * 4, wb = (nB * 9 + rp) * 4;
    int   ia0 = idx4[wa], ia1 = idx4[wa + 1], ia2 = idx4[wa + 2], ia3 = idx4[wa + 3];
    float ga0 = w4[wa],   ga1 = w4[wa + 1],   ga2 = w4[wa + 2],   ga3 = w4[wa + 3];
    int   ib0 = idx4[wb], ib1 = idx4[wb + 1], ib2 = idx4[wb + 2], ib3 = idx4[wb + 3];
    float gb0 = w4[wb],   gb1 = w4[wb + 1],   gb2 = w4[wb + 2],   gb3 = w4[wb + 3];
    for (int c0 = 0; c0 < 512; c0 += 4) {
      int c = c0 + 2 * lh;
      const float* p0 = cc + (size_t)c * NS;
      const float* p1 = cc + (size_t)(c + 1) * NS;
      v2f a0, a1, b0, b1;
      a0.x = Am0[c * 9 + rp];       a0.y = Am0[(c + 1) * 9 + rp];
      a1.x = Am1[c * 9 + rp];       a1.y = Am1[(c + 1) * 9 + rp];
      b0.x = ga0 * p0[ia0] + ga1 * p0[ia1] + ga2 * p0[ia2] + ga3 * p0[ia3];
      b0.y = ga0 * p1[ia0] + ga1 * p1[ia1] + ga2 * p1[ia2] + ga3 * p1[ia3];
      b1.x = gb0 * p0[ib0] + gb1 * p0[ib1] + gb2 * p0[ib2] + gb3 * p0[ib3];
      b1.y = gb0 * p1[ib0] + gb1 * p1[ib1] + gb2 * p1[ib2] + gb3 * p1[ib3];
      acc00 = wmma_f32(a0, b0, acc00);
      acc01 = wmma_f32(a0, b1, acc01);
      acc10 = wmma_f32(a1, b0, acc10);
      acc11 = wmma_f32(a1, b1, acc11);
    }
  }
#pragma unroll
  for (int r = 0; r < 8; ++r) {
    int mA = m0 + r + 8 * lh, mB = mA + 16;
    fam[(size_t)mA * NS + nA] = acc00[r];
    fam[(size_t)mA * NS + nB] = acc01[r];
    fam[(size_t)mB * NS + nA] = acc10[r];
    fam[(size_t)mB * NS + nB] = acc11[r];
  }
}

// ----------------------------------- 3x3 conv (256->256) + BN + ReLU, fused
// implicit im2col GEMM, K = 256*9, 2x2 wave tiles (32x32 outputs per wave)
__global__ void k_conv3bnrelu(const float* __restrict__ X, const float* __restrict__ W9,
                              const float* __restrict__ gamma, const float* __restrict__ beta,
                              const float* __restrict__ mean, const float* __restrict__ var,
                              float* __restrict__ O) {
  int lane = threadIdx.x & 31, wv = threadIdx.x >> 5;
  int t = blockIdx.x * 8 + wv;                       // 8 mt * 512 nt
  int mt = t >> 9, nt = t & 511;
  int m0 = mt * 32, n0 = nt * 32;
  int ln = lane & 15, lh = lane >> 4;
  int nA = n0 + ln,      hA = nA >> 7, wA = nA & 127;
  int nB = n0 + 16 + ln, hB = nB >> 7, wB = nB & 127;
  const float* Ar0 = W9 + (size_t)(m0 + ln) * 2304;
  const float* Ar1 = W9 + (size_t)(m0 + 16 + ln) * 2304;
  v8f acc00 = {}, acc01 = {}, acc10 = {}, acc11 = {};
  for (int k = 0; k < 2304; k += 4) {
    int ka = k + 2 * lh;
    v2f a0, a1, b0, b1;
    a0.x = Ar0[ka]; a0.y = Ar0[ka + 1];
    a1.x = Ar1[ka]; a1.y = Ar1[ka + 1];
#pragma unroll
    for (int vv = 0; vv < 2; ++vv) {
      int kk = ka + vv, c = kk / 9, r = kk - c * 9;
      int dy = r / 3 - 1, dx = (r % 3) - 1;
      int hhA = hA + dy, wwA = wA + dx;
      int hhB = hB + dy, wwB = wB + dx;
      float bvA = 0.f, bvB = 0.f;
      if (hhA >= 0 && hhA < 128 && wwA >= 0 && wwA < 128)
        bvA = X[(size_t)c * NS + hhA * 128 + wwA];
      if (hhB >= 0 && hhB < 128 && wwB >= 0 && wwB < 128)
        bvB = X[(size_t)c * NS + hhB * 128 + wwB];
      if (vv == 0) { b0.x = bvA; b1.x = bvB; }
      else         { b0.y = bvA; b1.y = bvB; }
    }
    acc00 = wmma_f32(a0, b0, acc00);
    acc01 = wmma_f32(a0, b1, acc01);
    acc10 = wmma_f32(a1, b0, acc10);
    acc11 = wmma_f32(a1, b1, acc11);
  }
#pragma unroll
  for (int r = 0; r < 8; ++r) {
    int mA = m0 + r + 8 * lh, mB = mA + 16;
    float sA = gamma[mA] * rsqrtf(var[mA] + 1e-5f);
    float sB = gamma[mB] * rsqrtf(var[mB] + 1e-5f);
    float cAo = beta[mA] - mean[mA] * sA;
    float cBo = beta[mB] - mean[mB] * sB;
    O[(size_t)mA * NS + nA] = fmaxf(acc00[r] * sA + cAo, 0.f);
    O[(size_t)mA * NS + nB] = fmaxf(acc01[r] * sA + cAo, 0.f);
    O[(size_t)mB * NS + nA] = fmaxf(acc10[r] * sB + cBo, 0.f);
    O[(size_t)mB * NS + nB] = fmaxf(acc11[r] * sB + cBo, 0.f);
  }
}

extern "C" void kernel_launch(void* const* d_in, const int* in_sizes, int n_in,
                              void* d_out, int out_size, void* d_ws, size_t ws_size,
                              hipStream_t stream) {
  const float* x   = (const float*)d_in[0];
  const float* L   = (const float*)d_in[1];
  const float* fm  = (const float*)d_in[2];
  const float* fs  = (const float*)d_in[3];
  const float* pw  = (const float*)d_in[4];
  const float* pb  = (const float*)d_in[5];
  const float* dcw = (const float*)d_in[6];
  const float* lc1 = (const float*)d_in[7];
  const float* g1  = (const float*)d_in[8];
  const float* b1  = (const float*)d_in[9];
  const float* m1  = (const float*)d_in[10];
  const float* v1  = (const float*)d_in[11];
  const float* lc2 = (const float*)d_in[12];
  const float* g2  = (const float*)d_in[13];
  const float* b2  = (const float*)d_in[14];
  const float* m2  = (const float*)d_in[15];
  const float* v2  = (const float*)d_in[16];

  float* out1 = (float*)d_out;
  float* out2 = out1 + (size_t)C1 * NS;

  float* ws  = (float*)d_ws;                 // ~90 MB of fp32 scratch
  float* cc  = ws;                           // [512][NS]: xu | Cf
  float* vv  = ws + 8388608;                 // [256][NS]
  float* uu  = ws + 12582912;                // [256][NS]
  float* off = ws + 16777216;                // [18][NS]
  int*   idx4= (int*)(ws + 17072128);        // [NS][9][4]
  float* w4  = ws + 17661952;                // [NS][9][4]
  float* fam = ws + 18251776;                // [256][NS]

  dim3 blk(256);
  k_upsample   <<<(C1 * NS) / 256, blk, 0, stream>>>(x, cc);
  k_gemm1x1    <<<512, blk, 0, stream>>>(fm, L, vv, 1);           // v = sigmoid(fm_w @ L)
  k_chanmm     <<<512, blk, 0, stream>>>(L, vv, uu);              // u = L@v + L (per ch)
  k_gemm1x1    <<<512, blk, 0, stream>>>(fs, uu, cc + (size_t)C1 * NS, 0); // Cf into cc
  k_offconv    <<<256, blk, 0, stream>>>(cc, pw, pb, off);
  k_dprep      <<<(NS * 9 + 255) / 256, blk, 0, stream>>>(off, idx4, w4);
  k_deform     <<<512, blk, 0, stream>>>(cc, dcw, idx4, w4, fam);
  k_conv3bnrelu<<<512, blk, 0, stream>>>(fam,  lc1, g1, b1, m1, v1, out1);
  k_conv3bnrelu<<<512, blk, 0, stream>>>(out1, lc2, g2, b2, m2, v2, out2);
}